// OneDynamicWeightsCat11_28097676050583
// MI455X (gfx1250) — compile-verified
//
#include <hip/hip_runtime.h>
#include <hip/hip_bf16.h>

// ---------------------------------------------------------------------------
// OneDynamicWeightsCat11 fused pipeline for gfx1250 (MI455X, wave32, WMMA).
// fp32 end-to-end, GEMMs use V_WMMA_F32_16X16X4_F32.
// ---------------------------------------------------------------------------

typedef float v2f __attribute__((ext_vector_type(2)));
typedef float v8f __attribute__((ext_vector_type(8)));

#define GN_GROUPS 32
#define EPS_GN 1e-5f

// Problem geometry is fixed by the reference; bake the pixel stride so global
// loads/stores use immediate offsets instead of per-iteration 64-bit address
// math (row stride 16384*4 B = 65536 B fits the signed 24-bit ioffset).
constexpr int kHW = 16384;      // H*W = 128*128

// ---------------------------------------------------------------------------
// WMMA fp32 GEMM:  Y[m,p] = sum_k W[m,k] * X[k,p] (+ bias[m])
// X is [K, kHW] row-major per image (channel-major NCHW slice), split-K over
// two sources (for the concat in scale2).  One wave computes a 16x64 tile via
// 4 accumulators.  Block = 256 threads = 8 waves arranged 2 (M) x 4 (P):
//   block tile = 32 rows x 256 cols.   grid = (kHW/256, M/32, N)
// A fragment (16x4 f32): lanes 0-15 rows M, VGPR0/1 = K {0,1}; lanes 16-31
// K {2,3}.  B mirrors it with N=lane.  D per ISA 16x16 f32 C/D layout.
// ---------------------------------------------------------------------------
__global__ __launch_bounds__(256) void gemm_wmma_f32_kernel(
    const float* __restrict__ Wmat,  // [M, K0+K1] row-major
    const float* __restrict__ X0,    // batched [N][K0][kHW]
    const float* __restrict__ X1,    // batched [N][K1][kHW] or nullptr
    const float* __restrict__ bias,  // [M] or nullptr
    float* __restrict__ Y,           // batched [N][M][kHW]
    int M, int K0, int K1)
{
    const int n    = blockIdx.z;
    const int lane = threadIdx.x & 31;
    const int wave = threadIdx.x >> 5;
    const int waveM = wave >> 2;   // 0..1
    const int waveP = wave & 3;    // 0..3
    const int Ktot = K0 + K1;

    const int m0 = (blockIdx.y * 2 + waveM) * 16;   // tile rows [m0, m0+16)
    const int p0 = (blockIdx.x * 4 + waveP) * 64;   // tile cols [p0, p0+64)

    const int half = lane >> 4;     // 0: K rows {0,1}; 1: K rows {2,3}
    const int l16  = lane & 15;

    // A fragment pointer: advances 4 floats per K-step across both K loops.
    const float* __restrict__ ap = Wmat + (long)(m0 + l16) * Ktot + half * 2;

    v8f acc0 = {0.f,0.f,0.f,0.f,0.f,0.f,0.f,0.f};
    v8f acc1 = acc0, acc2 = acc0, acc3 = acc0;

    // ---- K loop over X0 -------------------------------------------------
    {
        const float* __restrict__ xk =
            X0 + ((long)n * K0 + half * 2) * kHW + p0 + l16;
#pragma unroll 2
        for (int k = 0; k < K0; k += 4) {
            v2f a = *(const v2f*)ap;  ap += 4;
            v2f b0, b1, b2, b3;
            b0.x = xk[0];   b0.y = xk[kHW];
            b1.x = xk[16];  b1.y = xk[kHW + 16];
            b2.x = xk[32];  b2.y = xk[kHW + 32];
            b3.x = xk[48];  b3.y = xk[kHW + 48];
            xk += 4 * (long)kHW;
            acc0 = __builtin_amdgcn_wmma_f32_16x16x4_f32(false, a, false, b0, (short)0, acc0, false, false);
            acc1 = __builtin_amdgcn_wmma_f32_16x16x4_f32(false, a, false, b1, (short)0, acc1, false, false);
            acc2 = __builtin_amdgcn_wmma_f32_16x16x4_f32(false, a, false, b2, (short)0, acc2, false, false);
            acc3 = __builtin_amdgcn_wmma_f32_16x16x4_f32(false, a, false, b3, (short)0, acc3, false, false);
        }
    }
    // ---- K loop over X1 (concat tail) -----------------------------------
    if (X1) {
        const float* __restrict__ xk =
            X1 + ((long)n * K1 + half * 2) * kHW + p0 + l16;
#pragma unroll 2
        for (int k = 0; k < K1; k += 4) {
            v2f a = *(const v2f*)ap;  ap += 4;
            v2f b0, b1, b2, b3;
            b0.x = xk[0];   b0.y = xk[kHW];
            b1.x = xk[16];  b1.y = xk[kHW + 16];
            b2.x = xk[32];  b2.y = xk[kHW + 32];
            b3.x = xk[48];  b3.y = xk[kHW + 48];
            xk += 4 * (long)kHW;
            acc0 = __builtin_amdgcn_wmma_f32_16x16x4_f32(false, a, false, b0, (short)0, acc0, false, false);
            acc1 = __builtin_amdgcn_wmma_f32_16x16x4_f32(false, a, false, b1, (short)0, acc1, false, false);
            acc2 = __builtin_amdgcn_wmma_f32_16x16x4_f32(false, a, false, b2, (short)0, acc2, false, false);
            acc3 = __builtin_amdgcn_wmma_f32_16x16x4_f32(false, a, false, b3, (short)0, acc3, false, false);
        }
    }

    // bias per output row handled by this lane: m0 + half*8 + v
    float bv[8];
#pragma unroll
    for (int v = 0; v < 8; ++v)
        bv[v] = bias ? bias[m0 + half * 8 + v] : 0.0f;

    // D layout 16x16 fp32: lane<16 -> M=v, N=lane; lane>=16 -> M=v+8, N=lane-16
    float* __restrict__ yp =
        Y + ((long)n * M + m0 + half * 8) * kHW + p0 + l16;
    v8f accs[4] = {acc0, acc1, acc2, acc3};
#pragma unroll
    for (int t = 0; t < 4; ++t) {
#pragma unroll
        for (int v = 0; v < 8; ++v)
            yp[(long)v * kHW + t * 16] = accs[t][v] + bv[v];
    }
}

// ---------------------------------------------------------------------------
// GroupNorm statistics: one block per (n, group); float4 loads; writes
// stats[(n*32+g)*2 + {0,1}] = {mu, rsqrt(var+eps)}
// ---------------------------------------------------------------------------
__global__ __launch_bounds__(256) void gn_stats_kernel(
    const float* __restrict__ Y, float* __restrict__ stats,
    int C, int chPerGroup, float eps)
{
    const int n = blockIdx.x >> 5;
    const int g = blockIdx.x & 31;
    const float4* __restrict__ src =
        (const float4*)(Y + ((long)n * C + (long)g * chPerGroup) * kHW);
    const long cnt4 = (long)chPerGroup * kHW / 4;

    float s = 0.f, ss = 0.f;
    for (long i = threadIdx.x; i < cnt4; i += blockDim.x) {
        float4 v = src[i];
        s  += v.x + v.y + v.z + v.w;
        ss += v.x * v.x + v.y * v.y + v.z * v.z + v.w * v.w;
    }
    __shared__ float sh[256], sh2[256];
    sh[threadIdx.x] = s; sh2[threadIdx.x] = ss;
    __syncthreads();
    for (int ofs = 128; ofs > 0; ofs >>= 1) {
        if (threadIdx.x < ofs) {
            sh[threadIdx.x]  += sh[threadIdx.x + ofs];
            sh2[threadIdx.x] += sh2[threadIdx.x + ofs];
        }
        __syncthreads();
    }
    if (threadIdx.x == 0) {
        const float inv = 1.0f / (float)(cnt4 * 4);
        float mu  = sh[0] * inv;
        float var = sh2[0] * inv - mu * mu;
        stats[blockIdx.x * 2 + 0] = mu;
        stats[blockIdx.x * 2 + 1] = rsqrtf(var + eps);
    }
}

// ---------------------------------------------------------------------------
// GroupNorm apply (+ReLU, optional residual), in place, float4 per thread.
// kHW % 4 == 0, so all 4 elements of a float4 share one channel.
// ---------------------------------------------------------------------------
__global__ __launch_bounds__(256) void gn_apply_kernel(
    float* __restrict__ Y, const float* __restrict__ stats,
    const float* __restrict__ gamma, const float* __restrict__ beta,
    const float* __restrict__ resid,
    int C, int chPerGroup, long total4)
{
    const long i4 = (long)blockIdx.x * blockDim.x + threadIdx.x;
    if (i4 >= total4) return;
    const long idx = i4 * 4;
    const long c = (idx / kHW) % C;
    const long n = idx / ((long)kHW * C);
    const int  g = (int)(c / chPerGroup);
    const float mu = stats[(n * 32 + g) * 2 + 0];
    const float rs = stats[(n * 32 + g) * 2 + 1];
    const float sc = rs * gamma[c];
    const float sh = beta[c] - mu * sc;

    float4 v = *(const float4*)(Y + idx);
    v.x = fmaxf(fmaf(v.x, sc, sh), 0.0f);
    v.y = fmaxf(fmaf(v.y, sc, sh), 0.0f);
    v.z = fmaxf(fmaf(v.z, sc, sh), 0.0f);
    v.w = fmaxf(fmaf(v.w, sc, sh), 0.0f);
    if (resid) {
        float4 r = *(const float4*)(resid + idx);
        v.x += r.x; v.y += r.y; v.z += r.z; v.w += r.w;
    }
    *(float4*)(Y + idx) = v;
}

// ---------------------------------------------------------------------------
// Fused "cata" stage: 3x3 conv -> 9 logits/pixel -> softmax -> dynamic 3x3
// filtering of xd.  Block = 16x16 pixel tile; per-channel 18x18 halo tile and
// the 81 weights for that channel are staged in LDS.  filt stays in registers.
// grid = (W/16, H/16, N)
// ---------------------------------------------------------------------------
__global__ __launch_bounds__(256) void cata_fused_kernel(
    const float* __restrict__ xd,   // [N][Cr][H][W]
    const float* __restrict__ wc,   // [9][Cr][3][3]
    float* __restrict__ out1,       // [N][Cr][H][W]
    int Cr, int H, int Wd)
{
    __shared__ float tile[18 * 18];
    __shared__ float wsm[81];

    const int n  = blockIdx.z;
    const int tx = threadIdx.x & 15;
    const int ty = threadIdx.x >> 4;
    const int x0 = blockIdx.x * 16;
    const int y0 = blockIdx.y * 16;
    const int gx = x0 + tx;
    const int gy = y0 + ty;

    float logits[9];
#pragma unroll
    for (int k = 0; k < 9; ++k) logits[k] = 0.0f;

    // ---- pass 1: accumulate conv logits over channels ----
    for (int c = 0; c < Cr; ++c) {
        const float* __restrict__ src = xd + ((long)n * Cr + c) * kHW;
        for (int i = threadIdx.x; i < 18 * 18; i += 256) {
            int yy = i / 18 - 1 + y0;
            int xx = i % 18 - 1 + x0;
            tile[i] = (yy >= 0 && yy < H && xx >= 0 && xx < Wd) ? src[(long)yy * Wd + xx] : 0.0f;
        }
        if (threadIdx.x < 81) {
            int k = threadIdx.x / 9, e = threadIdx.x % 9;
            wsm[k * 9 + e] = wc[((long)k * Cr + c) * 9 + e];
        }
        __syncthreads();
#pragma unroll
        for (int k = 0; k < 9; ++k) {
            float a = 0.0f;
#pragma unroll
            for (int di = 0; di < 3; ++di)
#pragma unroll
                for (int dj = 0; dj < 3; ++dj)
                    a = fmaf(wsm[k * 9 + di * 3 + dj], tile[(ty + di) * 18 + tx + dj], a);
            logits[k] += a;
        }
        __syncthreads();
    }

    // ---- softmax over the 9 kernel positions (per pixel, in registers) ----
    float mx = logits[0];
#pragma unroll
    for (int k = 1; k < 9; ++k) mx = fmaxf(mx, logits[k]);
    float sum = 0.0f;
#pragma unroll
    for (int k = 0; k < 9; ++k) { logits[k] = __expf(logits[k] - mx); sum += logits[k]; }
    const float inv = 1.0f / sum;
#pragma unroll
    for (int k = 0; k < 9; ++k) logits[k] *= inv;   // logits[] now holds filt[]

    // ---- pass 2: dynamic filtering ----
    for (int c = 0; c < Cr; ++c) {
        const float* __restrict__ src = xd + ((long)n * Cr + c) * kHW;
        for (int i = threadIdx.x; i < 18 * 18; i += 256) {
            int yy = i / 18 - 1 + y0;
            int xx = i % 18 - 1 + x0;
            tile[i] = (yy >= 0 && yy < H && xx >= 0 && xx < Wd) ? src[(long)yy * Wd + xx] : 0.0f;
        }
        __syncthreads();
        float a = 0.0f;
#pragma unroll
        for (int di = 0; di < 3; ++di)
#pragma unroll
            for (int dj = 0; dj < 3; ++dj)
                a = fmaf(logits[di * 3 + dj], tile[(ty + di) * 18 + tx + dj], a);
        out1[((long)n * Cr + c) * kHW + (long)gy * Wd + gx] = a;
        __syncthreads();
    }
}

// ---------------------------------------------------------------------------
// Host-side launch sequence
// ---------------------------------------------------------------------------
extern "C" void kernel_launch(void* const* d_in, const int* in_sizes, int n_in,
                              void* d_out, int out_size, void* d_ws, size_t ws_size,
                              hipStream_t stream)
{
    const int N = 4, C = 512, Cr = 128, H = 128, Wd = 128;
    const long szCr = (long)N * Cr * kHW;          // 8,388,608 elements
    const long szC  = (long)N * C  * kHW;          // 33,554,432 elements

    const float* x   = (const float*)d_in[0];
    const float* w1  = (const float*)d_in[1];
    const float* g1  = (const float*)d_in[2];
    const float* b1  = (const float*)d_in[3];
    const float* wc  = (const float*)d_in[4];
    const float* w2  = (const float*)d_in[5];
    const float* bw2 = (const float*)d_in[6];
    const float* g2  = (const float*)d_in[7];
    const float* b2  = (const float*)d_in[8];
    const float* w3  = (const float*)d_in[9];
    const float* bw3 = (const float*)d_in[10];
    const float* g3  = (const float*)d_in[11];
    const float* b3  = (const float*)d_in[12];
    float* out = (float*)d_out;

    // workspace layout
    float* xd    = (float*)d_ws;       // [N][Cr][kHW]  (scale1 pre -> xd in place)
    float* out1  = xd   + szCr;        // [N][Cr][kHW]
    float* h2    = out1 + szCr;        // [N][Cr][kHW]  (scale2 pre -> h2 in place)
    float* stats = h2   + szCr;        // 256 floats

    const dim3 blk(256, 1, 1);

    // ---- scale1: xd_pre = w1 . x ----------------------------------------
    {
        dim3 grid(kHW / 256, Cr / 32, N);
        gemm_wmma_f32_kernel<<<grid, blk, 0, stream>>>(
            w1, x, nullptr, nullptr, xd, Cr, C, 0);
    }
    gn_stats_kernel<<<dim3(N * GN_GROUPS), blk, 0, stream>>>(
        xd, stats, Cr, Cr / GN_GROUPS, EPS_GN);
    gn_apply_kernel<<<dim3((unsigned)((szCr / 4 + 255) / 256)), blk, 0, stream>>>(
        xd, stats, g1, b1, nullptr, Cr, Cr / GN_GROUPS, szCr / 4);

    // ---- cata: conv3x3 -> softmax -> dynamic filtering -> out1 ----------
    {
        dim3 grid(Wd / 16, H / 16, N);
        cata_fused_kernel<<<grid, blk, 0, stream>>>(xd, wc, out1, Cr, H, Wd);
    }

    // ---- scale2: h2_pre = w2 . concat(xd, out1) + bw2 -------------------
    {
        dim3 grid(kHW / 256, Cr / 32, N);
        gemm_wmma_f32_kernel<<<grid, blk, 0, stream>>>(
            w2, xd, out1, bw2, h2, Cr, Cr, Cr);
    }
    gn_stats_kernel<<<dim3(N * GN_GROUPS), blk, 0, stream>>>(
        h2, stats, Cr, Cr / GN_GROUPS, EPS_GN);
    gn_apply_kernel<<<dim3((unsigned)((szCr / 4 + 255) / 256)), blk, 0, stream>>>(
        h2, stats, g2, b2, nullptr, Cr, Cr / GN_GROUPS, szCr / 4);

    // ---- scale3: h3_pre = w3 . h2 + bw3  (written straight to d_out) ----
    {
        dim3 grid(kHW / 256, C / 32, N);
        gemm_wmma_f32_kernel<<<grid, blk, 0, stream>>>(
            w3, h2, nullptr, bw3, out, C, Cr, 0);
    }
    gn_stats_kernel<<<dim3(N * GN_GROUPS), blk, 0, stream>>>(
        out, stats, C, C / GN_GROUPS, EPS_GN);
    // final GN + ReLU + residual add, in place on d_out
    gn_apply_kernel<<<dim3((unsigned)((szC / 4 + 255) / 256)), blk, 0, stream>>>(
        out, stats, g3, b3, x, C, C / GN_GROUPS, szC / 4);
}